// Net_88381837017215
// MI455X (gfx1250) — compile-verified
//
#include <hip/hip_runtime.h>

#define N_NODES 100000
#define N_EDGES 1600000
#define IN_F 256
#define HID 128
#define OUT 64

typedef __attribute__((ext_vector_type(2))) float v2f;
typedef __attribute__((ext_vector_type(8))) float v8f;

// ---------------------------------------------------------------------------
// GEMM: C[M x N] = op(A)[M x K] @ W[K x N] using V_WMMA_F32_16X16X4_F32.
// One wave computes one 16x16 output tile; wave index selects the N-tile,
// blockIdx selects the M-tile. M = 100000 = 6250 * 16 exactly, so EXEC is
// always all-ones (WMMA requirement).
//
// A layout (16x4 f32, 2 VGPRs/lane): lanes 0-15 hold M=0..15 with K={0,1},
// lanes 16-31 hold M=0..15 with K={2,3}.
// B layout (4x16 f32, 2 VGPRs/lane): lanes 0-15 hold N=0..15 rows K={0,1},
// lanes 16-31 hold N=0..15 rows K={2,3}.
// C/D layout (16x16 f32, 8 VGPRs): vgpr v, lanes 0-15 -> (M=v, N=lane),
// lanes 16-31 -> (M=v+8, N=lane-16).
// ---------------------------------------------------------------------------
template <int K, int N, bool RELU_A>
__global__ void gemm_wmma_f32(const float* __restrict__ A,
                              const float* __restrict__ W,
                              float* __restrict__ C) {
    const int lane  = threadIdx.x & 31;
    const int wave  = threadIdx.x >> 5;     // N-tile index within row of tiles
    const int row0  = blockIdx.x * 16;      // M-tile base
    const int col0  = wave * 16;            // N-tile base
    const int l15   = lane & 15;
    const int khalf = (lane >> 4) * 2;      // 0 for lanes 0-15, 2 for 16-31

    const float* __restrict__ Arow = A + (size_t)(row0 + l15) * K;
    const float* __restrict__ Wcol = W + col0 + l15;

    v8f acc = {};
    for (int k0 = 0; k0 < K; k0 += 4) {
        const int ka = k0 + khalf;
        float a0 = Arow[ka];
        float a1 = Arow[ka + 1];
        if (RELU_A) { a0 = fmaxf(a0, 0.0f); a1 = fmaxf(a1, 0.0f); }
        v2f a; a.x = a0; a.y = a1;
        v2f b; b.x = Wcol[(size_t)ka * N];
               b.y = Wcol[(size_t)(ka + 1) * N];
        // (neg_a, A, neg_b, B, c_mod, C, reuse_a, reuse_b)
        acc = __builtin_amdgcn_wmma_f32_16x16x4_f32(false, a, false, b,
                                                    (short)0, acc, false, false);
    }

    const int rowc = row0 + (lane >> 4) * 8;    // lanes 16-31 hold rows M+8
    float* __restrict__ Cp = C + (size_t)rowc * N + col0 + l15;
#pragma unroll
    for (int v = 0; v < 8; ++v) {
        Cp[(size_t)v * N] = acc[v];
    }
}

// ---------------------------------------------------------------------------
// Zero a float buffer (scatter targets must start at 0; ws is poisoned 0xAA).
// ---------------------------------------------------------------------------
__global__ void zero_f32(float* __restrict__ p, size_t n) {
    size_t i      = (size_t)blockIdx.x * blockDim.x + threadIdx.x;
    size_t stride = (size_t)gridDim.x * blockDim.x;
    for (; i < n; i += stride) p[i] = 0.0f;
}

// ---------------------------------------------------------------------------
// Edge-parallel SpMM scatter: Y[row[e],:] += val[e] * X[col[e],:].
// D/4 threads cooperate on one edge (float4 chunks) so the gather of a row
// is fully coalesced; intermediates live in the 192MB L2, so the random
// gathers/atomics stay on-chip. unsafeAtomicAdd -> native GLOBAL_ATOMIC_ADD_F32.
// ---------------------------------------------------------------------------
template <int D>
__global__ void spmm_scatter(const int* __restrict__ erow,
                             const int* __restrict__ ecol,
                             const float* __restrict__ eval_,
                             const float* __restrict__ X,
                             float* __restrict__ Y) {
    constexpr int CPE = D / 4;  // float4 chunks per edge (power of two)
    const size_t total  = (size_t)N_EDGES * CPE;
    const size_t stride = (size_t)gridDim.x * blockDim.x;
    for (size_t i = (size_t)blockIdx.x * blockDim.x + threadIdx.x;
         i < total; i += stride) {
        const size_t e = i / CPE;
        const int    c = (int)(i & (CPE - 1)) * 4;
        const int    r = erow[e];
        const int    s = ecol[e];
        const float  v = eval_[e];
        const float4 xv = *(const float4*)(X + (size_t)s * D + c);
        float* __restrict__ yp = Y + (size_t)r * D + c;
        unsafeAtomicAdd(yp + 0, v * xv.x);
        unsafeAtomicAdd(yp + 1, v * xv.y);
        unsafeAtomicAdd(yp + 2, v * xv.z);
        unsafeAtomicAdd(yp + 3, v * xv.w);
    }
}

// ---------------------------------------------------------------------------
// Fused relu + row softmax over 64 columns. One wave32 per row; each lane
// owns columns {lane, lane+32}; reductions via __shfl_xor (wave32-safe).
// ---------------------------------------------------------------------------
__global__ void relu_softmax64(const float* __restrict__ X,
                               float* __restrict__ Y) {
    const int lane = threadIdx.x & 31;
    const int wave = threadIdx.x >> 5;
    const int row  = blockIdx.x * (blockDim.x >> 5) + wave;
    if (row >= N_NODES) return;

    const float* __restrict__ x = X + (size_t)row * OUT;
    float a = fmaxf(x[lane], 0.0f);
    float b = fmaxf(x[lane + 32], 0.0f);

    float m = fmaxf(a, b);
#pragma unroll
    for (int off = 16; off > 0; off >>= 1)
        m = fmaxf(m, __shfl_xor(m, off, 32));

    const float ea = __expf(a - m);
    const float eb = __expf(b - m);
    float s = ea + eb;
#pragma unroll
    for (int off = 16; off > 0; off >>= 1)
        s += __shfl_xor(s, off, 32);

    const float inv = 1.0f / s;
    float* __restrict__ y = Y + (size_t)row * OUT;
    y[lane]      = ea * inv;
    y[lane + 32] = eb * inv;
}

// ---------------------------------------------------------------------------
// Pipeline: H1 = X@W1 ; AGG = A*H1 ; H2 = relu(AGG)@W2 ; AGG2 = A*H2 ;
// out = softmax(relu(AGG2)). Workspace regions reused: peak 2*N*HID*4 = 102MB.
// ---------------------------------------------------------------------------
extern "C" void kernel_launch(void* const* d_in, const int* in_sizes, int n_in,
                              void* d_out, int out_size, void* d_ws, size_t ws_size,
                              hipStream_t stream) {
    const float* features = (const float*)d_in[0];
    const int*   erow     = (const int*)d_in[1];
    const int*   ecol     = (const int*)d_in[2];
    const float* eval_    = (const float*)d_in[3];
    const float* W1       = (const float*)d_in[4];
    const float* W2       = (const float*)d_in[5];
    float*       out      = (float*)d_out;

    float* H1   = (float*)d_ws;                    // [N, HID]
    float* AGG  = H1 + (size_t)N_NODES * HID;      // [N, HID]
    float* H2   = H1;                              // [N, OUT]  (reuse region A)
    float* AGG2 = AGG;                             // [N, OUT]  (reuse region B)

    const int MT = N_NODES / 16;                   // 6250 M-tiles (exact)

    // 1) H1 = X @ W1   (8 waves/block cover HID=128 tile columns)
    gemm_wmma_f32<IN_F, HID, false><<<MT, 256, 0, stream>>>(features, W1, H1);

    // 2) AGG = A * H1
    zero_f32<<<2048, 256, 0, stream>>>(AGG, (size_t)N_NODES * HID);
    spmm_scatter<HID><<<8192, 256, 0, stream>>>(erow, ecol, eval_, H1, AGG);

    // 3) H2 = relu(AGG) @ W2   (4 waves/block cover OUT=64 tile columns)
    gemm_wmma_f32<HID, OUT, true><<<MT, 128, 0, stream>>>(AGG, W2, H2);

    // 4) AGG2 = A * H2
    zero_f32<<<2048, 256, 0, stream>>>(AGG2, (size_t)N_NODES * OUT);
    spmm_scatter<OUT><<<8192, 256, 0, stream>>>(erow, ecol, eval_, H2, AGG2);

    // 5) out = softmax(relu(AGG2), dim=-1)
    relu_softmax64<<<(N_NODES + 7) / 8, 256, 0, stream>>>(AGG2, out);
}